// SphConvAttention_89910845374840
// MI455X (gfx1250) — compile-verified
//
#include <hip/hip_runtime.h>
#include <hip/hip_bf16.h>

typedef __attribute__((ext_vector_type(2))) float v2f;
typedef __attribute__((ext_vector_type(4))) float v4f;
typedef __attribute__((ext_vector_type(8))) float v8f;

#define HIDDEN 128
#define M_TOT 16

// ---------------- Kernel 0: zero the output (poisoned by harness) ----------
__global__ void zero_kernel(float* __restrict__ out, int n) {
    int i = blockIdx.x * blockDim.x + threadIdx.x;
    if (i < n) out[i] = 0.0f;
}

// ---------------- Kernel 1: Q = X*Wq^T, K = X*Wk^T via WMMA f32 16x16x4 ----
// Block: 256 threads = 8 waves. Each block owns a 16-row tile of X (in LDS).
// Wave w computes the 16-column tile [16w,16w+16) of both Q and K.
__global__ void __launch_bounds__(256)
qk_gemm_kernel(const float* __restrict__ X,
               const float* __restrict__ Wq,
               const float* __restrict__ Wk,
               float* __restrict__ Q,
               float* __restrict__ K) {
    __shared__ float xs[16][HIDDEN];
    const int m0 = blockIdx.x * 16;

    // Cooperative load of 16x128 X tile (2048 floats) as float4
    {
        const float4* src = (const float4*)(X + (size_t)m0 * HIDDEN);
        float4* dst = (float4*)&xs[0][0];
        dst[threadIdx.x]       = src[threadIdx.x];
        dst[threadIdx.x + 256] = src[threadIdx.x + 256];
    }
    __syncthreads();

    const int waveId = threadIdx.x >> 5;
    const int lane   = threadIdx.x & 31;
    const int ntile  = waveId * 16;
    const int row    = lane & 15;     // M for A-frag, N for B/C-frag
    const int hi     = lane >> 4;     // lane-half select

    v8f cq = {};
    v8f ck = {};

    for (int kk = 0; kk < HIDDEN; kk += 4) {
        // A-frag (16x4 f32): lanes 0-15 -> K = kk+{0,1}; lanes 16-31 -> K = kk+{2,3}
        v2f a;
        a.x = xs[row][kk + hi * 2 + 0];
        a.y = xs[row][kk + hi * 2 + 1];
        // B-frag (4x16): B[k][n] = W[n][k]; VGPR j: lanes 0-15 -> K=kk+2*hi+j
        const int n  = ntile + row;
        const int kb = kk + hi * 2;
        v2f bq, bk;
        bq.x = Wq[n * HIDDEN + kb + 0];
        bq.y = Wq[n * HIDDEN + kb + 1];
        bk.x = Wk[n * HIDDEN + kb + 0];
        bk.y = Wk[n * HIDDEN + kb + 1];

        cq = __builtin_amdgcn_wmma_f32_16x16x4_f32(false, a, false, bq,
                                                   (short)0, cq, false, false);
        ck = __builtin_amdgcn_wmma_f32_16x16x4_f32(false, a, false, bk,
                                                   (short)0, ck, false, false);
    }

    // C/D layout: VGPR v, lanes 0-15 -> M=v, lanes 16-31 -> M=v+8; N = lane&15
    const int col   = ntile + row;
    const int rbase = m0 + hi * 8;
#pragma unroll
    for (int v = 0; v < 8; ++v) {
        Q[(size_t)(rbase + v) * HIDDEN + col] = cq[v];
        K[(size_t)(rbase + v) * HIDDEN + col] = ck[v];
    }
}

// ---------------- Kernel 2: per-edge attention + scatter -------------------
// One wave32 per edge. Lane l loads float4 of q_i/k_j/w at element 4l;
// elements [4l,4l+3] all belong to head (l>>3). 8-lane butterfly reduction
// gives each head's dot product; lanes 0-15 expand per REPEATS=[1,3,5,7] and
// atomically accumulate alpha*sph into chi_out[dst].
//
// L2 policy: w_ij / sph_ij are streamed once -> non-temporal loads (TH=NT),
// so the 51 MB Q/K gather set + 3.2 MB atomic output stay resident in the
// 192 MB L2. src/dst are wave-uniform -> readfirstlane for SGPR gather bases.
__global__ void __launch_bounds__(256)
edge_kernel(const float* __restrict__ Q,
            const float* __restrict__ K,
            const float* __restrict__ w_ij,
            const float* __restrict__ sph,
            const int*   __restrict__ ei,
            const float* __restrict__ cutoff,
            float* __restrict__ out,
            int nEdges) {
    const int e    = (blockIdx.x * blockDim.x + threadIdx.x) >> 5;
    const int lane = threadIdx.x & 31;
    if (e >= nEdges) return;

    // wave-uniform edge metadata -> scalar registers
    const int src = __builtin_amdgcn_readfirstlane(ei[e]);          // edge_index[0][e]
    const int dst = __builtin_amdgcn_readfirstlane(ei[nEdges + e]); // edge_index[1][e]

    const v4f qa = ((const v4f*)(Q + (size_t)dst * HIDDEN))[lane];
    const v4f ka = ((const v4f*)(K + (size_t)src * HIDDEN))[lane];
    const v4f wa = __builtin_nontemporal_load(
        (const v4f*)(w_ij + (size_t)e * HIDDEN) + lane);

    float p = qa.x * wa.x * ka.x + qa.y * wa.y * ka.y +
              qa.z * wa.z * ka.z + qa.w * wa.w * ka.w;

    // reduce within each group of 8 lanes (one head per group)
    p += __shfl_xor(p, 1, 32);
    p += __shfl_xor(p, 2, 32);
    p += __shfl_xor(p, 4, 32);

    const float scale = cutoff[e] * 0.17677669529663689f; // 1/sqrt(32)
    const float a0 = __shfl(p, 0, 32) * scale;
    const float a1 = __shfl(p, 8, 32) * scale;
    const float a2 = __shfl(p, 16, 32) * scale;
    const float a3 = __shfl(p, 24, 32) * scale;

    if (lane < M_TOT) {
        // repeat map over [1,3,5,7]: m=0->h0, 1..3->h1, 4..8->h2, 9..15->h3
        float alpha = (lane < 1) ? a0 : (lane < 4) ? a1 : (lane < 9) ? a2 : a3;
        float s = __builtin_nontemporal_load(sph + (size_t)e * M_TOT + lane);
        atomicAdd(&out[(size_t)dst * M_TOT + lane], alpha * s);
    }
}

// ---------------------------------------------------------------------------
extern "C" void kernel_launch(void* const* d_in, const int* in_sizes, int n_in,
                              void* d_out, int out_size, void* d_ws, size_t ws_size,
                              hipStream_t stream) {
    // inputs: 0 chi, 1 sph_ij, 2 x, 3 w_ij, 4 edge_index, 5 cutoff, 6 Wq, 7 Wk
    const float* sph  = (const float*)d_in[1];
    const float* x    = (const float*)d_in[2];
    const float* w_ij = (const float*)d_in[3];
    const int*   ei   = (const int*)d_in[4];
    const float* cut  = (const float*)d_in[5];
    const float* Wq   = (const float*)d_in[6];
    const float* Wk   = (const float*)d_in[7];
    float* out = (float*)d_out;

    const int nEdges = in_sizes[4] / 2;
    const int nNodes = in_sizes[2] / HIDDEN;

    float* Q = (float*)d_ws;
    float* K = Q + (size_t)nNodes * HIDDEN;

    zero_kernel<<<(out_size + 255) / 256, 256, 0, stream>>>(out, out_size);
    qk_gemm_kernel<<<nNodes / 16, 256, 0, stream>>>(x, Wq, Wk, Q, K);
    edge_kernel<<<(nEdges + 7) / 8, 256, 0, stream>>>(Q, K, w_ij, sph, ei, cut,
                                                      out, nEdges);
}